// SelectiveROIAlign_49039936586127
// MI455X (gfx1250) — compile-verified
//
#include <hip/hip_runtime.h>

// ---------------------------------------------------------------------------
// SelectiveROIAlign for MI455X (gfx1250).
// Bottleneck: one streaming pass over 32MB of scores (~1.4us at 23.3 TB/s),
// streamed via the CDNA5 Tensor Data Mover (tensor_load_to_lds, double-
// buffered LDS tiles + s_wait_tensorcnt). Exact distributed top-k.
// ROI bilinear crop is factored as W[7x28] @ G[28x256] per (box,level,gy)
// and run on V_WMMA_F32_16X16X4_F32 (layouts per ISA 7.12.2).
// ---------------------------------------------------------------------------

#define MAXP        10
#define TK          20
#define LSTRIDE     21      // coprime with 64 banks -> conflict-free lists
#define K1_BLOCKS   256
#define K1_THREADS  256
#define CHUNK       2048    // score elements per TDM tile (8KB)
#define SCORE_THR   0.05f
#define GS          260     // padded LDS stride for G rows

#if defined(__gfx1250__) && __has_builtin(__builtin_amdgcn_tensor_load_to_lds)
#define USE_TDM 1
#else
#define USE_TDM 0
#endif

#if __has_include(<hip/amd_detail/amd_gfx1250_TDM.h>)
#define TDM_6ARG 1
#else
#define TDM_6ARG 0
#endif

typedef unsigned int u32x4 __attribute__((ext_vector_type(4)));
typedef int          i32x4 __attribute__((ext_vector_type(4)));
typedef int          i32x8 __attribute__((ext_vector_type(8)));
typedef float        v2f   __attribute__((ext_vector_type(2)));
typedef float        v8f   __attribute__((ext_vector_type(8)));

// ------------------------- CDNA5 wait helpers ------------------------------
__device__ __forceinline__ void wait_tensorcnt0() {
#if USE_TDM
#if __has_builtin(__builtin_amdgcn_s_wait_tensorcnt)
  __builtin_amdgcn_s_wait_tensorcnt(0);
#else
  asm volatile("s_wait_tensorcnt 0x0" ::: "memory");
#endif
#endif
}
__device__ __forceinline__ void wait_tensorcnt1() {
#if USE_TDM
#if __has_builtin(__builtin_amdgcn_s_wait_tensorcnt)
  __builtin_amdgcn_s_wait_tensorcnt(1);
#else
  asm volatile("s_wait_tensorcnt 0x1" ::: "memory");
#endif
#endif
}

// ------------------------- TDM tile issue ----------------------------------
// D# packing per CDNA5 ISA (count=1, data_size=4B,
// 1-row tile; tensor_dim0 = remaining elements so tail tiles zero-fill OOB).
#if USE_TDM
__device__ __forceinline__ void tdm_load_f32_tile(const float* src, unsigned ldsByteOff,
                                                  unsigned long long tensorElems,
                                                  unsigned tileElems) {
  unsigned long long ga = (unsigned long long)(const void*)src;
  unsigned td0 = (unsigned)(tensorElems > 0xFFFFFFFFull ? 0xFFFFFFFFull : tensorElems);
  unsigned long long s0 = (unsigned long long)td0;

  u32x4 g0;
  g0.x = 1u;                                                   // count=1, user D#
  g0.y = ldsByteOff;                                           // lds_addr
  g0.z = (unsigned)(ga & 0xFFFFFFFFull);                       // global_addr[31:0]
  g0.w = (unsigned)((ga >> 32) & 0x01FFFFFFull) | (2u << 30);  // addr[56:32] | type=2

  i32x8 g1;
  g1[0] = (int)(2u << 16);                                     // data_size = 4 bytes
  g1[1] = (int)((td0 & 0xFFFFu) << 16);                        // tensor_dim0[15:0]
  g1[2] = (int)(((td0 >> 16) & 0xFFFFu) | (1u << 16));         // td0[31:16] | td1=1
  g1[3] = (int)((tileElems & 0xFFFFu) << 16);                  // tile_dim0
  g1[4] = (int)1;                                              // tile_dim1=1
  g1[5] = (int)(s0 & 0xFFFFFFFFull);                           // dim0_stride[31:0]
  g1[6] = (int)((s0 >> 32) & 0xFFFFull);                       // dim0_stride[47:32]
  g1[7] = 0;

  i32x4 g2 = {0, 0, 0, 0};
  i32x4 g3 = {0, 0, 0, 0};
#if TDM_6ARG
  i32x8 g4 = {0, 0, 0, 0, 0, 0, 0, 0};
  __builtin_amdgcn_tensor_load_to_lds(g0, g1, g2, g3, g4, 0);
#else
  __builtin_amdgcn_tensor_load_to_lds(g0, g1, g2, g3, 0);
#endif
}
#endif // USE_TDM

// ------------------------- top-k primitives --------------------------------
__device__ __forceinline__ bool tk_better(float v1, int i1, float v2, int i2) {
  return (v1 > v2) || (v1 == v2 && i1 < i2);   // lax.top_k tie-break: lower index
}

__device__ __forceinline__ void tk_insert(float* lv, int* li, int& cnt, float nv, int ni) {
  if (cnt == TK) {
    if (!tk_better(nv, ni, lv[TK - 1], li[TK - 1])) return;
    cnt = TK - 1;
  }
  int p = cnt;
  while (p > 0 && tk_better(nv, ni, lv[p - 1], li[p - 1])) {
    lv[p] = lv[p - 1]; li[p] = li[p - 1]; --p;
  }
  lv[p] = nv; li[p] = ni; ++cnt;
}

__device__ void block_merge20(float* lval, int* lidx, int cnt,
                              float* rv, int* ri, int* rt, int* wsel,
                              float* outv, int* outi) {
  const int tid = threadIdx.x;
  int hp = 0;
  for (int r = 0; r < TK; ++r) {
    float mv = -3.0f; int mi = 0x7FFFFFFF;
    if (hp < cnt) { mv = lval[tid * LSTRIDE + hp]; mi = lidx[tid * LSTRIDE + hp]; }
    rv[tid] = mv; ri[tid] = mi; rt[tid] = tid;
    __syncthreads();
    for (int s = K1_THREADS >> 1; s > 0; s >>= 1) {
      if (tid < s) {
        if (tk_better(rv[tid + s], ri[tid + s], rv[tid], ri[tid])) {
          rv[tid] = rv[tid + s]; ri[tid] = ri[tid + s]; rt[tid] = rt[tid + s];
        }
      }
      __syncthreads();
    }
    if (tid == 0) { outv[r] = rv[0]; outi[r] = ri[0]; *wsel = rt[0]; }
    __syncthreads();
    if (tid == *wsel) ++hp;
    __syncthreads();
  }
}

// ------------------------- kernel 1: partial top-k -------------------------
__global__ __launch_bounds__(K1_THREADS)
void k_topk_partial(const float* __restrict__ cls, long long NE,
                    float* __restrict__ outS, int* __restrict__ outI) {
  __shared__ float lval[K1_THREADS * LSTRIDE];
  __shared__ int   lidx[K1_THREADS * LSTRIDE];
  __shared__ float rv[K1_THREADS];
  __shared__ int   ri[K1_THREADS];
  __shared__ int   rt[K1_THREADS];
  __shared__ float outv[TK];
  __shared__ int   outi[TK];
  __shared__ int   wsel;
#if USE_TDM
  __shared__ float stage[2][CHUNK];
#endif

  const int tid = threadIdx.x;
  const long long per = (NE + K1_BLOCKS - 1) / K1_BLOCKS;
  const int  tiles = (int)((per + CHUNK - 1) / CHUNK);
  const long long blockStart = (long long)blockIdx.x * (long long)tiles * CHUNK;
  int cnt = 0;

#if USE_TDM
  const int wave = tid >> 5;
  if (wave == 0) {
    long long rem = NE - blockStart;
    tdm_load_f32_tile(cls + (rem > 0 ? blockStart : 0),
                      (unsigned)(unsigned long long)&stage[0][0],
                      (unsigned long long)(rem > 0 ? rem : 1), CHUNK);
  }
  for (int t = 0; t < tiles; ++t) {
    if (wave == 0) {
      if (t + 1 < tiles) {
        long long off = blockStart + (long long)(t + 1) * CHUNK;
        long long rem = NE - off;
        tdm_load_f32_tile(cls + (rem > 0 ? off : 0),
                          (unsigned)(unsigned long long)&stage[(t + 1) & 1][0],
                          (unsigned long long)(rem > 0 ? rem : 1), CHUNK);
        wait_tensorcnt1();                 // tile t complete (in-order per wave)
      } else {
        wait_tensorcnt0();
      }
    }
    __syncthreads();
    const float* buf = stage[t & 1];
    const long long base = blockStart + (long long)t * CHUNK;
    for (int k = tid; k < CHUNK; k += K1_THREADS) {
      long long g = base + k;
      if (g < NE) {
        float v = buf[k];
        float m = (v > SCORE_THR) ? v : -1.0f;
        tk_insert(&lval[tid * LSTRIDE], &lidx[tid * LSTRIDE], cnt, m, (int)g);
      }
    }
    __syncthreads();
  }
#else
  const long long span = (long long)tiles * CHUNK;
  for (long long k = tid; k < span; k += K1_THREADS) {
    long long g = blockStart + k;
    if (g < NE) {
      float v = cls[g];
      float m = (v > SCORE_THR) ? v : -1.0f;
      tk_insert(&lval[tid * LSTRIDE], &lidx[tid * LSTRIDE], cnt, m, (int)g);
    }
  }
  __syncthreads();
#endif

  block_merge20(lval, lidx, cnt, rv, ri, rt, &wsel, outv, outi);
  if (tid < TK) {
    outS[blockIdx.x * TK + tid] = outv[tid];
    outI[blockIdx.x * TK + tid] = outi[tid];
  }
}

// ------------------------- kernel 2: final select --------------------------
__global__ __launch_bounds__(K1_THREADS)
void k_topk_final(const float* __restrict__ inS, const int* __restrict__ inI,
                  const float* __restrict__ boxes, int nbox, int Cn,
                  float* __restrict__ selBox, int* __restrict__ selMask) {
  __shared__ float lval[K1_THREADS * LSTRIDE];
  __shared__ int   lidx[K1_THREADS * LSTRIDE];
  __shared__ float rv[K1_THREADS];
  __shared__ int   ri[K1_THREADS];
  __shared__ int   rt[K1_THREADS];
  __shared__ float outv[TK];
  __shared__ int   outi[TK];
  __shared__ int   wsel;

  const int tid = threadIdx.x;
  const int NC = K1_BLOCKS * TK;
  int cnt = 0;
  for (int j = tid; j < NC; j += K1_THREADS)
    tk_insert(&lval[tid * LSTRIDE], &lidx[tid * LSTRIDE], cnt, inS[j], inI[j]);
  __syncthreads();

  block_merge20(lval, lidx, cnt, rv, ri, rt, &wsel, outv, outi);

  if (tid == 0) {
    int uniq[MAXP]; int nk = 0;
    int seen[TK];  int ns = 0;
    for (int j = 0; j < TK; ++j) {
      float s = outv[j];
      if (!(s > SCORE_THR)) continue;
      int a = outi[j] / Cn;
      bool first = true;
      for (int q = 0; q < ns; ++q) if (seen[q] == a) { first = false; break; }
      seen[ns++] = a;
      if (first && nk < MAXP) uniq[nk++] = a;
    }
    for (int s2 = 0; s2 < MAXP; ++s2) {
      int a = (s2 < nk) ? uniq[s2] : 0;
      a = a < 0 ? 0 : (a >= nbox ? nbox - 1 : a);
      const float* b = boxes + (long long)a * 4;
      selBox[s2 * 4 + 0] = b[1];     // y1
      selBox[s2 * 4 + 1] = b[0];     // x1
      selBox[s2 * 4 + 2] = b[3];     // y2
      selBox[s2 * 4 + 3] = b[2];     // x2
      selMask[s2] = (s2 < nk) ? 1 : 0;
    }
  }
}

// ------------------------- kernel 3: WMMA ROI align ------------------------
// For one (box,level,gy): pooled[gy][gx 0..6][c 0..255] = W[7x28] @ G[28x256]
//   K index k = a*14 + j ; a in {0,1} selects {y0,y1i}; j = 2*gx + b selects
//   {x0,x1i}. W[gx][k] = valid * wy_a * wx_b  (zero rows reproduce the
//   reference's per-level mask before the cross-level max).
// V_WMMA_F32_16X16X4_F32 layouts (ISA 7.12.2):
//   A 16x4 : lane L, M=L&15, VGPR0/1 = K (kb,kb+1), kb = 2*(L>>4)
//   B 4x16 : lane L, N=L&15, VGPR0/1 = K (kb,kb+1)
//   C 16x16: lane L, VGPR r -> row r + 8*(L>>4), col L&15
__global__ __launch_bounds__(256)
void k_roi_wmma(const float* __restrict__ f0, const float* __restrict__ f1,
                const float* __restrict__ f2, const float* __restrict__ f3,
                const float* __restrict__ selBox, float* __restrict__ pooled) {
  __shared__ float Gl[28 * GS];
  __shared__ float Wl[16 * 28];

  const int bl  = blockIdx.x;           // box*4 + level
  const int box = bl >> 2, lvl = bl & 3;
  const int tid = threadIdx.x;
  const int lane = tid & 31, wv = tid >> 5;
  const int HS[4] = {256, 128, 64, 32};
  const int H = HS[lvl], Wd = H;
  const float Hm1 = (float)(H - 1), Wm1 = (float)(Wd - 1);
  const float y1 = selBox[box * 4 + 0], x1 = selBox[box * 4 + 1];
  const float y2 = selBox[box * 4 + 2], x2 = selBox[box * 4 + 3];
  const float* img = (lvl == 0) ? f0 : (lvl == 1) ? f1 : (lvl == 2) ? f2 : f3;

  const int M  = lane & 15;
  const int kb = (lane >> 4) * 2;
  const int n0 = (wv * 2) * 16 + (lane & 15);
  const int n1 = n0 + 16;

  for (int gy = 0; gy < 7; ++gy) {
    const float in_y = y1 * Hm1 + (float)gy * ((y2 - y1) * Hm1 / 6.0f);
    const bool vy = (in_y >= 0.0f && in_y <= Hm1);
    float fy = floorf(in_y); fy = fy < 0.0f ? 0.0f : (fy > Hm1 ? Hm1 : fy);
    const int y0  = (int)fy;
    const int y1i = (y0 + 1 < H) ? y0 + 1 : H - 1;
    const float wy = in_y - (float)y0;

    for (int i = tid; i < 16 * 28; i += 256) Wl[i] = 0.0f;
    __syncthreads();
    if (tid < 7) {
      const int gx = tid;
      const float in_x = x1 * Wm1 + (float)gx * ((x2 - x1) * Wm1 / 6.0f);
      const bool vx = (in_x >= 0.0f && in_x <= Wm1);
      float fx = floorf(in_x); fx = fx < 0.0f ? 0.0f : (fx > Wm1 ? Wm1 : fx);
      const int x0 = (int)fx;
      const float wx = in_x - (float)x0;
      const float valid = (vy && vx) ? 1.0f : 0.0f;
      Wl[gx * 28 +  0 + 2 * gx + 0] = valid * (1.0f - wy) * (1.0f - wx);
      Wl[gx * 28 +  0 + 2 * gx + 1] = valid * (1.0f - wy) * wx;
      Wl[gx * 28 + 14 + 2 * gx + 0] = valid * wy * (1.0f - wx);
      Wl[gx * 28 + 14 + 2 * gx + 1] = valid * wy * wx;
    }
    // stage G[28][256]: gathered feature rows (clipped indices, like reference)
    for (int k = 0; k < 28; ++k) {
      const int a = k / 14, j = k % 14;
      const int gx = j >> 1, bsel = j & 1;
      const float in_x = x1 * Wm1 + (float)gx * ((x2 - x1) * Wm1 / 6.0f);
      float fx = floorf(in_x); fx = fx < 0.0f ? 0.0f : (fx > Wm1 ? Wm1 : fx);
      const int x0 = (int)fx;
      const int xx = bsel ? ((x0 + 1 < Wd) ? x0 + 1 : Wd - 1) : x0;
      const int yy = a ? y1i : y0;
      Gl[k * GS + tid] = img[((long long)yy * Wd + xx) * 256 + tid];
    }
    __syncthreads();

    v8f c0 = {}; v8f c1 = {};
#pragma unroll
    for (int s = 0; s < 7; ++s) {
      const int k4 = s * 4;
      v2f a, b0, b1;
      a.x  = Wl[M * 28 + k4 + kb];
      a.y  = Wl[M * 28 + k4 + kb + 1];
      b0.x = Gl[(k4 + kb) * GS + n0];
      b0.y = Gl[(k4 + kb + 1) * GS + n0];
      b1.x = Gl[(k4 + kb) * GS + n1];
      b1.y = Gl[(k4 + kb + 1) * GS + n1];
      c0 = __builtin_amdgcn_wmma_f32_16x16x4_f32(false, a, false, b0, (short)0, c0, false, false);
      c1 = __builtin_amdgcn_wmma_f32_16x16x4_f32(false, a, false, b1, (short)0, c1, false, false);
    }

    float* dst = pooled + ((long long)bl * 49 + gy * 7) * 256;
#pragma unroll
    for (int r = 0; r < 8; ++r) {
      const int row = r + 8 * (lane >> 4);
      if (row < 7) {
        dst[(long long)row * 256 + n0] = c0[r];
        dst[(long long)row * 256 + n1] = c1[r];
      }
    }
    __syncthreads();
  }
}

// ------------------------- kernel 4: level max + mask ----------------------
__global__ __launch_bounds__(256)
void k_roi_max(const float* __restrict__ pooled, const int* __restrict__ selMask,
               float* __restrict__ out) {
  const int b = blockIdx.x, ch = threadIdx.x;
  const int m = selMask[b];
  for (int p = 0; p < 49; ++p) {
    const float v0 = pooled[((long long)(b * 4 + 0) * 49 + p) * 256 + ch];
    const float v1 = pooled[((long long)(b * 4 + 1) * 49 + p) * 256 + ch];
    const float v2 = pooled[((long long)(b * 4 + 2) * 49 + p) * 256 + ch];
    const float v3 = pooled[((long long)(b * 4 + 3) * 49 + p) * 256 + ch];
    const float fu = fmaxf(fmaxf(v0, v1), fmaxf(v2, v3));
    out[((long long)b * 49 + p) * 256 + ch] = m ? fu : 0.0f;
  }
}

// ------------------------- host launch -------------------------------------
extern "C" void kernel_launch(void* const* d_in, const int* in_sizes, int n_in,
                              void* d_out, int out_size, void* d_ws, size_t ws_size,
                              hipStream_t stream) {
  const float* f0    = (const float*)d_in[0];
  const float* f1    = (const float*)d_in[1];
  const float* f2    = (const float*)d_in[2];
  const float* f3    = (const float*)d_in[3];
  const float* boxes = (const float*)d_in[4];
  const float* cls   = (const float*)d_in[5];

  const long long NE = (long long)in_sizes[5];        // 100000*80 = 8e6
  const int nbox = in_sizes[4] / 4;                   // 100000
  const int Cn   = (int)(NE / (long long)nbox);       // 80

  char* ws = (char*)d_ws;
  float* pS      = (float*)ws;                                    // [K1_BLOCKS*TK]
  int*   pI      = (int*)(ws + (size_t)K1_BLOCKS * TK * 4);       // [K1_BLOCKS*TK]
  float* selBox  = (float*)(ws + (size_t)2 * K1_BLOCKS * TK * 4); // [MAXP*4]
  int*   selMask = (int*)(ws + (size_t)2 * K1_BLOCKS * TK * 4 + (size_t)MAXP * 4 * 4);
  float* pooled  = (float*)(ws + (size_t)65536);                  // [40*49*256] ~2MB

  k_topk_partial<<<K1_BLOCKS, K1_THREADS, 0, stream>>>(cls, NE, pS, pI);
  k_topk_final<<<1, K1_THREADS, 0, stream>>>(pS, pI, boxes, nbox, Cn, selBox, selMask);
  k_roi_wmma<<<MAXP * 4, 256, 0, stream>>>(f0, f1, f2, f3, selBox, pooled);
  k_roi_max<<<MAXP, 256, 0, stream>>>(pooled, selMask, (float*)d_out);
}